// GINLayer_38491496907253
// MI455X (gfx1250) — compile-verified
//
#include <hip/hip_runtime.h>

typedef float v2f __attribute__((ext_vector_type(2)));
typedef float v8f __attribute__((ext_vector_type(8)));
typedef unsigned int u32x4 __attribute__((ext_vector_type(4)));
typedef int i32x4 __attribute__((ext_vector_type(4)));
typedef int i32x8 __attribute__((ext_vector_type(8)));

#define N_NODES 40000
#define N_EDGES 640000
#define DIM 128
#define KSTEPS (DIM / 4)        // 32 WMMA K-steps per GEMM
#define X_STRIDE 132            // X tile row stride (floats): 128 + 4 pad
// Repacked W: 64 "kp-rows" of 128 float2 pairs (256 DW); TDM pads 32 DW/row
// -> LDS kp-row stride = 288 DW = 144 float2 (288 mod 64 banks = 32: the
//    hi half-wave (kp-row + 1) hits the opposite 32-bank half; conflict-free)
#define WP_ROWS (DIM / 2)               // 64 kp-rows
#define WP_ROW_DW 256                   // DWORDs per kp-row in memory
#define WP_LDS_STRIDE_PAIRS 144         // float2 stride per kp-row in LDS
#define WP_LDS_FLOATS (WP_ROWS * WP_LDS_STRIDE_PAIRS * 2)   // 18432
#define WAVES_PER_BLOCK 8
#define ROWS_PER_WAVE 16
#define ROWS_PER_BLOCK (WAVES_PER_BLOCK * ROWS_PER_WAVE)
#define SMEM_FLOATS (2 * WP_LDS_FLOATS + WAVES_PER_BLOCK * ROWS_PER_WAVE * X_STRIDE)
#define NEIGH_FLOATS ((size_t)N_NODES * DIM)
#define WP_FLOATS (DIM * DIM)           // 16384 floats per repacked matrix

// ---------------------------------------------------------------------------
// Edge scatter: neigh[dst] += h[src] * mask.  One wave32 per edge; each lane
// carries one float4 (32 lanes x 4 = 128 = D).  L2-atomic bound by design
// (h is 20.5 MB -> resident in the 192 MB L2, so gathers hit cache).
// ---------------------------------------------------------------------------
__global__ __launch_bounds__(256)
void gin_scatter(const float4* __restrict__ h4, const int* __restrict__ src,
                 const int* __restrict__ dst, const float* __restrict__ mask,
                 float* __restrict__ neigh) {
    const int gid  = blockIdx.x * blockDim.x + threadIdx.x;
    const int e    = gid >> 5;
    const int lane = threadIdx.x & 31;
    if (e >= N_EDGES) return;

    const int   s = src[e];
    const int   d = dst[e];
    const float m = mask[e];

    const float4 v = h4[(size_t)s * (DIM / 4) + lane];
    float* p = neigh + (size_t)d * DIM + lane * 4;
    atomicAdd(p + 0, v.x * m);
    atomicAdd(p + 1, v.y * m);
    atomicAdd(p + 2, v.z * m);
    atomicAdd(p + 3, v.w * m);
}

// ---------------------------------------------------------------------------
// One-time weight repack: WP[kp*128 + c] = { W[2kp][c], W[2kp+1][c] }.
// Makes a WMMA B-fragment (two consecutive K values, same column) a single
// contiguous 8-byte ds_load_b64 once staged in LDS.
// ---------------------------------------------------------------------------
__global__ __launch_bounds__(256)
void gin_repack_w(const float* __restrict__ W1, const float* __restrict__ W2,
                  float2* __restrict__ wp1, float2* __restrict__ wp2) {
    const int p = blockIdx.x * blockDim.x + threadIdx.x;  // pair index
    if (p >= (DIM / 2) * DIM) return;
    const int c  = p & (DIM - 1);
    const int kp = p >> 7;
    wp1[p] = make_float2(W1[(kp * 2) * DIM + c], W1[(kp * 2 + 1) * DIM + c]);
    wp2[p] = make_float2(W2[(kp * 2) * DIM + c], W2[(kp * 2 + 1) * DIM + c]);
}

// ---------------------------------------------------------------------------
// TDM bulk copy of one repacked weight matrix (64 rows x 256 DW, contiguous)
// into LDS, inserting 32 pad DWORDs after every 256 DWORDs (row stride 288).
// Descriptor per CDNA5 ISA.  EXEC ignored by TDM;
// completion tracked with TENSORcnt.
// ---------------------------------------------------------------------------
__device__ inline void tdm_load_wp(const float* __restrict__ gsrc, float* ldst) {
    const unsigned long long ga = (unsigned long long)(uintptr_t)gsrc;
    const unsigned lds_addr = (unsigned)(uintptr_t)ldst;  // flat low 32 = LDS byte addr

    u32x4 g0;
    g0.x = 1u;                                    // count=1 (valid user D#)
    g0.y = lds_addr;                              // lds_addr[31:0]
    g0.z = (unsigned)(ga & 0xFFFFFFFFu);          // global_addr[31:0]
    g0.w = (unsigned)((ga >> 32) & 0x01FFFFFFu)   // global_addr[56:32]
         | (2u << 30);                            // type = 2 ("image")

    i32x8 g1 = {};
    g1[0] = (2 << 16)          // data_size = 2 -> 4-byte elements
          | (1 << 20)          // pad_enable
          | (7 << 22)          // pad_interval: 2^(7+1) = 256 DWORDs
          | (31 << 25);        // pad_amount:   31+1    = 32 DWORDs
    g1[1] = (int)((unsigned)WP_ROW_DW << 16);  // tensor_dim0[15:0] (bits 63:48)
    g1[2] = (int)((unsigned)WP_ROWS << 16);    // tensor_dim1[15:0] (bits 95:80)
    g1[3] = (int)((unsigned)WP_ROW_DW << 16);  // tile_dim0 (bits 127:112)
    g1[4] = WP_ROWS;                           // tile_dim1 (bits 143:128)
    g1[5] = WP_ROW_DW;                         // tensor_dim0_stride (bits 191:160)
    g1[6] = 0;
    g1[7] = 0;

    i32x4 gz = {};             // groups 2/3 unused for 2D tensors
#if defined(__clang_major__) && (__clang_major__ >= 23)
    i32x8 gz8 = {};
    __builtin_amdgcn_tensor_load_to_lds(g0, g1, gz, gz, gz8, 0);
#else
    __builtin_amdgcn_tensor_load_to_lds(g0, g1, gz, gz, 0);
#endif
}

// ---------------------------------------------------------------------------
// One 16x128 @ 128x128 GEMM tile with software double-buffered fragments:
// loads for K-step k+1 are issued BEFORE the WMMA burst for K-step k, so the
// LDS round-trip hides behind 8 v_wmma_f32_16x16x4_f32.  The loop body is
// fully symmetric (prefetch index clamped on the final iteration instead of
// a peeled tail) so the back-edge phi is an identity and the allocator can
// ping-pong fragment buffers without accumulator copies.
// ---------------------------------------------------------------------------
__device__ __forceinline__ v2f ld_a(const float* X, int lo, int hi, int kk) {
    return *(const v2f*)(X + lo * X_STRIDE + kk * 4 + 2 * hi);
}
__device__ __forceinline__ void ld_b(const float* wlds, int lo, int hi, int kk,
                                     v2f bf[8]) {
    const v2f* wrow = (const v2f*)wlds + (2 * kk + hi) * WP_LDS_STRIDE_PAIRS + lo;
#pragma unroll
    for (int n = 0; n < 8; ++n) bf[n] = wrow[n * 16];
}
__device__ __forceinline__ void gemm_tile(const float* X, const float* wlds,
                                          int lo, int hi, v8f acc[8]) {
    v2f a0, a1, b0[8], b1[8];
    a0 = ld_a(X, lo, hi, 0);
    ld_b(wlds, lo, hi, 0, b0);
    for (int kk = 0; kk < KSTEPS; kk += 2) {
        const int k1 = kk + 1;                               // always < KSTEPS
        const int k2 = (kk + 2 < KSTEPS) ? kk + 2 : 0;       // clamped prefetch
        a1 = ld_a(X, lo, hi, k1);
        ld_b(wlds, lo, hi, k1, b1);
#pragma unroll
        for (int n = 0; n < 8; ++n)
            acc[n] = __builtin_amdgcn_wmma_f32_16x16x4_f32(
                false, a0, false, b0[n], (short)0, acc[n], false, false);
        a0 = ld_a(X, lo, hi, k2);
        ld_b(wlds, lo, hi, k2, b0);
#pragma unroll
        for (int n = 0; n < 8; ++n)
            acc[n] = __builtin_amdgcn_wmma_f32_16x16x4_f32(
                false, a1, false, b1[n], (short)0, acc[n], false, false);
    }
}

// ---------------------------------------------------------------------------
// Fused combine + 2-layer MLP + PReLU, all f32 via V_WMMA_F32_16X16X4_F32.
// Repacked W1/W2 are TDM-copied into LDS once per block; each wave owns a
// private 16x128 LDS X tile (LDS ops are in-order per wave -> no intra-wave
// barriers around the X tile).
//
// WMMA 16x16x4 f32 fragment layouts (ISA 7.12.2):
//   A (16x4):  lane<16 -> M=lane, {v0,v1}=K{0,1};  lane>=16 -> K{2,3}
//   B (4x16):  lane<16 -> N=lane, {v0,v1}=K{0,1};  lane>=16 -> K{2,3}
//   C (16x16): elem(M = v + 8*(lane>>4), N = 16n + (lane&15)) in acc[n][v]
// ---------------------------------------------------------------------------
__global__ __launch_bounds__(WAVES_PER_BLOCK * 32)
void gin_mlp(const float* __restrict__ h, const float* __restrict__ neigh,
             const float* __restrict__ epsp, const float* __restrict__ wp1,
             const float* __restrict__ b1, const float* __restrict__ wp2,
             const float* __restrict__ b2, const float* __restrict__ alphap,
             float* __restrict__ out) {
    extern __shared__ float smem[];
    float* w1s = smem;                       // 64 x 288 DW (repacked W1)
    float* w2s = smem + WP_LDS_FLOATS;       // 64 x 288 DW (repacked W2)
    float* xs  = smem + 2 * WP_LDS_FLOATS;   // 8 waves x 16 x 132

    const int tid  = threadIdx.x;
    const int w    = tid >> 5;
    const int lane = tid & 31;
    const int lo   = lane & 15;
    const int hi   = lane >> 4;
    const int rowbase = blockIdx.x * ROWS_PER_BLOCK + w * ROWS_PER_WAVE;

    // ---- kick off TDM weight staging (waves 0 and 1) ----------------------
    if (w == 0)      tdm_load_wp(wp1, w1s);
    else if (w == 1) tdm_load_wp(wp2, w2s);

    const float scale = 1.0f + epsp[0];
    const float slope = alphap[0];

    float* X = xs + w * (ROWS_PER_WAVE * X_STRIDE);

    // ---- stage h_new = (1+eps)*h + neigh (overlaps TDM in flight) ---------
    for (int r = 0; r < ROWS_PER_WAVE; ++r) {
        const int row = rowbase + r;
        float4 v = make_float4(0.f, 0.f, 0.f, 0.f);
        if (row < N_NODES) {
            const float4 hv = ((const float4*)h)[(size_t)row * (DIM / 4) + lane];
            const float4 nv = ((const float4*)neigh)[(size_t)row * (DIM / 4) + lane];
            v.x = scale * hv.x + nv.x;
            v.y = scale * hv.y + nv.y;
            v.z = scale * hv.z + nv.z;
            v.w = scale * hv.w + nv.w;
        }
        *(float4*)(X + r * X_STRIDE + lane * 4) = v;
    }

    // issuing waves drain their TENSORcnt, then everyone syncs on weights
    if (w < 2) __builtin_amdgcn_s_wait_tensorcnt(0);
    __syncthreads();

    v8f acc[8];

    // ---- layer 1: Y = relu(X @ W1 + b1) -----------------------------------
    for (int n = 0; n < 8; ++n) { v8f z = {}; acc[n] = z; }
    gemm_tile(X, w1s, lo, hi, acc);
    // relu + bias, write back over this wave's X tile (all X reads done)
    for (int n = 0; n < 8; ++n) {
        const float bias = b1[n * 16 + lo];
#pragma unroll
        for (int v = 0; v < 8; ++v) {
            float y = acc[n][v] + bias;
            y = (y > 0.f) ? y : 0.f;
            X[(v + 8 * hi) * X_STRIDE + n * 16 + lo] = y;
        }
    }

    // ---- layer 2: Z = Y @ W2 + b2, then PReLU -----------------------------
    for (int n = 0; n < 8; ++n) { v8f z = {}; acc[n] = z; }
    gemm_tile(X, w2s, lo, hi, acc);
    for (int n = 0; n < 8; ++n) {
        const float bias = b2[n * 16 + lo];
#pragma unroll
        for (int v = 0; v < 8; ++v) {
            const int row = rowbase + v + 8 * hi;
            if (row < N_NODES) {
                float y = acc[n][v] + bias;
                y = (y >= 0.f) ? y : slope * y;
                out[(size_t)row * DIM + n * 16 + lo] = y;
            }
        }
    }
}

// ---------------------------------------------------------------------------
extern "C" void kernel_launch(void* const* d_in, const int* in_sizes, int n_in,
                              void* d_out, int out_size, void* d_ws, size_t ws_size,
                              hipStream_t stream) {
    const float* h     = (const float*)d_in[0];
    // d_in[1] = snorm_n (unused by reference)
    const int*   src   = (const int*)d_in[2];
    const int*   dst   = (const int*)d_in[3];
    const float* mask  = (const float*)d_in[4];
    const float* eps   = (const float*)d_in[5];
    const float* W1    = (const float*)d_in[6];
    const float* b1    = (const float*)d_in[7];
    const float* W2    = (const float*)d_in[8];
    const float* b2    = (const float*)d_in[9];
    const float* alpha = (const float*)d_in[10];
    float*       out   = (float*)d_out;

    float* neigh = (float*)d_ws;                  // N_NODES*DIM f32 accumulator
    float* wp1   = (float*)d_ws + NEIGH_FLOATS;   // repacked W1 (64 KB)
    float* wp2   = wp1 + WP_FLOATS;               // repacked W2 (64 KB)

    hipMemsetAsync(neigh, 0, NEIGH_FLOATS * sizeof(float), stream);

    {   // one-time weight repack into K-pair-interleaved layout
        const int blocks = ((DIM / 2) * DIM + 255) / 256;   // 32
        gin_repack_w<<<blocks, 256, 0, stream>>>(
            W1, W2, (float2*)wp1, (float2*)wp2);
    }
    {   // one wave per edge
        const long long threads = (long long)N_EDGES * 32;
        const int blocks = (int)((threads + 255) / 256);
        gin_scatter<<<blocks, 256, 0, stream>>>(
            (const float4*)h, src, dst, mask, neigh);
    }
    {   // 128 rows per block (8 waves x 16-row tiles)
        const int blocks = (N_NODES + ROWS_PER_BLOCK - 1) / ROWS_PER_BLOCK;
        const size_t smem_bytes = (size_t)SMEM_FLOATS * sizeof(float);
        gin_mlp<<<blocks, WAVES_PER_BLOCK * 32, smem_bytes, stream>>>(
            h, neigh, eps, wp1, b1, wp2, b2, alpha, out);
    }
}